// TemporalSelfAttention_17927193494094
// MI455X (gfx1250) — compile-verified
//
#include <hip/hip_runtime.h>
#include <hip/hip_bf16.h>
#include <math.h>

// ---------------------------------------------------------------------------
// TemporalSelfAttention (BEVFormer) for MI455X / gfx1250, wave32 + WMMA f32.
//
// Shapes (from reference): bs=2, nq=16384, c=256, H=8 heads, hd=32,
// Q(bev queue)=2, P(points)=4, level=1 of 128x128.
// ---------------------------------------------------------------------------

#define BS   2
#define NQ   16384
#define C    256
#define NH   8
#define HD   32
#define NP   4
#define NQUE 2
#define IMGH 128
#define IMGW 128

typedef __attribute__((ext_vector_type(2))) float v2f;
typedef __attribute__((ext_vector_type(8))) float v8f;

// ---------------------------------------------------------------------------
// Kernel 1: v = query @ Wv + bv, written as value images [b][h][q][ch]
// (channel-innermost so the sampling kernel's corner loads are coalesced).
// grid = (NQ/32, 2, BS), block = 256 threads = 8 waves.
// Each wave computes TWO 16x16 M-tiles sharing one B fragment per K-step
// (2 WMMAs per pair of global B loads).
// ---------------------------------------------------------------------------
__global__ __launch_bounds__(256) void gemm_value_kernel(
    const float* __restrict__ query,   // (BS, NQ, C)
    const float* __restrict__ Wv,      // (C, C)
    const float* __restrict__ bv,      // (C)
    float* __restrict__ vimg)          // (BS, NH, NQ, HD)
{
    __shared__ float As[32 * 260];     // 32 rows x 256 K, stride 260 (pad 4)

    const int tid = threadIdx.x;
    const int q0  = blockIdx.x * 32;
    const int b   = blockIdx.z;

    // Stage the 32x256 A tile: thread t loads 32 contiguous floats of row t/8.
    {
        const int r  = tid >> 3;          // 0..31
        const int c0 = (tid & 7) * 32;    // 0,32,...,224
        const float* src = query + ((size_t)b * NQ + (q0 + r)) * C + c0;
        float* dst = &As[r * 260 + c0];
        #pragma unroll
        for (int i = 0; i < 32; i += 4) {
            float4 t = *(const float4*)(src + i);
            dst[i + 0] = t.x; dst[i + 1] = t.y; dst[i + 2] = t.z; dst[i + 3] = t.w;
        }
    }
    __syncthreads();

    const int wave = tid >> 5;
    const int lane = tid & 31;
    const int half = lane >> 4;          // 0 or 1
    const int col  = lane & 15;          // 0..15
    const int n    = blockIdx.y * 128 + wave * 16 + col;   // output channel

    // Bias folded into accumulators (bias depends only on N == lane's column).
    const float bias = bv[n];
    v8f acc0, acc1;
    #pragma unroll
    for (int i = 0; i < 8; ++i) { acc0[i] = bias; acc1[i] = bias; }

    const float* arow0 = &As[(col +  0) * 260 + 2 * half];  // tile0: M = col
    const float* arow1 = &As[(col + 16) * 260 + 2 * half];  // tile1: M = 16+col
    const float* bcol  = Wv + (size_t)(2 * half) * C + n;   // B[K=2*half+v][n]

    #pragma unroll 4
    for (int k = 0; k < C; k += 4) {
        // Prefetch B rows a few K-steps ahead (global_prefetch_b8).
        __builtin_prefetch(bcol + (size_t)(k + 32) * C, 0, 1);

        v2f bb, a0, a1;
        bb.x = bcol[(size_t)k * C];
        bb.y = bcol[(size_t)(k + 1) * C];
        a0.x = arow0[k];  a0.y = arow0[k + 1];
        a1.x = arow1[k];  a1.y = arow1[k + 1];
        acc0 = __builtin_amdgcn_wmma_f32_16x16x4_f32(
            false, a0, false, bb, (short)0, acc0, false, false);
        acc1 = __builtin_amdgcn_wmma_f32_16x16x4_f32(
            false, a1, false, bb, (short)0, acc1, false, false);
    }

    // Scatter into [b][h][q][ch] image layout. D frag: M = 8*half+v, N = col.
    const int h  = n >> 5;        // head
    const int ch = n & 31;        // channel within head
    float* o0 = vimg + (((size_t)(b * NH + h)) * NQ + (q0 +      8 * half)) * HD + ch;
    float* o1 = vimg + (((size_t)(b * NH + h)) * NQ + (q0 + 16 + 8 * half)) * HD + ch;
    #pragma unroll
    for (int v = 0; v < 8; ++v) {
        o0[(size_t)v * HD] = acc0[v];
        o1[(size_t)v * HD] = acc1[v];
    }
}

// ---------------------------------------------------------------------------
// Kernel 2: raw off/aw = q_cat @ [Wo | Wa] + [bo | ba]  -> (BS, NQ, 192)
// q_cat[b][q] = concat(query[0][q], query[b][q] + query_pos[b][q])  (K = 512)
// grid = (NQ/32, BS), block = 384 threads = 12 waves (8 for Wo, 4 for Wa).
// K staged in two 256-wide halves through one 33 KB LDS buffer; each wave
// computes two M-tiles sharing B fragments.
// ---------------------------------------------------------------------------
__global__ __launch_bounds__(384) void gemm_offaw_kernel(
    const float* __restrict__ query,     // (BS, NQ, C)
    const float* __restrict__ query_pos, // (BS, NQ, C)
    const float* __restrict__ Wo,        // (512, 128)
    const float* __restrict__ bo,        // (128)
    const float* __restrict__ Wa,        // (512, 64)
    const float* __restrict__ ba,        // (64)
    float* __restrict__ offaw)           // (BS, NQ, 192)
{
    __shared__ float As[32 * 260];       // 32 rows x 256 K-half, stride 260

    const int tid = threadIdx.x;
    const int q0  = blockIdx.x * 32;
    const int b   = blockIdx.y;

    const int wave = tid >> 5;           // 0..11
    const int lane = tid & 31;
    const int half = lane >> 4;
    const int col  = lane & 15;

    const float* B;
    int ldN, n_local, n_global;
    float bias;
    if (wave < 8) {                       // Wo block: N = 128
        B = Wo; ldN = 128;
        n_local  = wave * 16 + col;
        n_global = n_local;
        bias = bo[n_local];
    } else {                              // Wa block: N = 64
        B = Wa; ldN = 64;
        n_local  = (wave - 8) * 16 + col;
        n_global = 128 + n_local;
        bias = ba[n_local];
    }

    v8f acc0, acc1;
    #pragma unroll
    for (int i = 0; i < 8; ++i) { acc0[i] = bias; acc1[i] = bias; }

    const float* arow0 = &As[(col +  0) * 260 + 2 * half];
    const float* arow1 = &As[(col + 16) * 260 + 2 * half];

    for (int kh = 0; kh < 2; ++kh) {
        __syncthreads();                  // WAR: previous pass done reading
        // Stage 32 rows x 256 cols of this K-half of q_cat.
        for (int i = tid; i < 32 * 256; i += 384) {
            const int r  = i >> 8;
            const int kk = i & 255;
            float v;
            if (kh == 0) {
                // reference quirk: value[:bs] rows are all query[0] for bs==2
                v = query[((size_t)0 * NQ + (q0 + r)) * C + kk];
            } else {
                const size_t idx = ((size_t)b * NQ + (q0 + r)) * C + kk;
                v = query[idx] + query_pos[idx];
            }
            As[r * 260 + kk] = v;
        }
        __syncthreads();

        const float* bcol = B + (size_t)(kh * 256 + 2 * half) * ldN + n_local;

        #pragma unroll 4
        for (int k = 0; k < 256; k += 4) {
            __builtin_prefetch(bcol + (size_t)(k + 32) * ldN, 0, 1);

            v2f bb, a0, a1;
            bb.x = bcol[(size_t)k * ldN];
            bb.y = bcol[(size_t)(k + 1) * ldN];
            a0.x = arow0[k];  a0.y = arow0[k + 1];
            a1.x = arow1[k];  a1.y = arow1[k + 1];
            acc0 = __builtin_amdgcn_wmma_f32_16x16x4_f32(
                false, a0, false, bb, (short)0, acc0, false, false);
            acc1 = __builtin_amdgcn_wmma_f32_16x16x4_f32(
                false, a1, false, bb, (short)0, acc1, false, false);
        }
    }

    float* o0 = offaw + ((size_t)(b * NQ + q0 +      8 * half)) * 192 + n_global;
    float* o1 = offaw + ((size_t)(b * NQ + q0 + 16 + 8 * half)) * 192 + n_global;
    #pragma unroll
    for (int v = 0; v < 8; ++v) {
        o0[(size_t)v * 192] = acc0[v];
        o1[(size_t)v * 192] = acc1[v];
    }
}

// ---------------------------------------------------------------------------
// Kernel 3: softmax over 4 points, bilinear gather (zeros padding), weighted
// sum, mean over BEV queue, + identity.  One wave per (b, q, h); lane = ch.
// grid = BS*NQ*NH/8 blocks, block = 256 threads = 8 waves.
// ---------------------------------------------------------------------------
__global__ __launch_bounds__(256) void sample_kernel(
    const float* __restrict__ query,   // (BS, NQ, C)   (identity)
    const float* __restrict__ refpts,  // (BS*NQUE, NQ, 1, 2)
    const float* __restrict__ vimg,    // (BS, NH, NQ=128*128, HD)
    const float* __restrict__ offaw,   // (BS, NQ, 192)
    float* __restrict__ out)           // (BS, NQ, C)
{
    const int lane = threadIdx.x & 31;         // channel within head
    const int wsub = threadIdx.x >> 5;
    const int w    = blockIdx.x * 8 + wsub;    // global wave id
    const int h    = w & (NH - 1);
    const int q    = (w / NH) & (NQ - 1);
    const int b    = w / (NH * NQ);

    const float* oa   = offaw + (size_t)(b * NQ + q) * 192;
    const float* img  = vimg + ((size_t)(b * NH + h)) * NQ * HD;

    float acc = 0.f;

    #pragma unroll
    for (int j = 0; j < NQUE; ++j) {
        // softmax over the 4 points of (h, j)
        const int abase = 128 + h * (NQUE * NP) + j * NP;
        float a0 = oa[abase + 0], a1 = oa[abase + 1];
        float a2 = oa[abase + 2], a3 = oa[abase + 3];
        float m  = fmaxf(fmaxf(a0, a1), fmaxf(a2, a3));
        float e0 = __expf(a0 - m), e1 = __expf(a1 - m);
        float e2 = __expf(a2 - m), e3 = __expf(a3 - m);
        float inv = 1.f / (e0 + e1 + e2 + e3);
        float wts[NP] = {e0 * inv, e1 * inv, e2 * inv, e3 * inv};

        const int b2 = b * NQUE + j;
        const float rx = refpts[((size_t)b2 * NQ + q) * 2 + 0];
        const float ry = refpts[((size_t)b2 * NQ + q) * 2 + 1];

        #pragma unroll
        for (int p = 0; p < NP; ++p) {
            const int obase = ((h * NQUE + j) * NP + p) * 2;
            const float ox = oa[obase + 0];
            const float oy = oa[obase + 1];
            // loc in [0,1]; pixel coords with align_corners=False
            const float x = (rx + ox * (1.f / IMGW)) * IMGW - 0.5f;
            const float y = (ry + oy * (1.f / IMGH)) * IMGH - 0.5f;
            const float xf = floorf(x), yf = floorf(y);
            const float fx = x - xf,  fy = y - yf;
            const int x0 = (int)xf, y0 = (int)yf;

            float s = 0.f;
            #pragma unroll
            for (int dy = 0; dy < 2; ++dy) {
                #pragma unroll
                for (int dx = 0; dx < 2; ++dx) {
                    const int xi = x0 + dx, yi = y0 + dy;
                    const float wx = dx ? fx : (1.f - fx);
                    const float wy = dy ? fy : (1.f - fy);
                    const bool valid = (xi >= 0) & (xi < IMGW) &
                                       (yi >= 0) & (yi < IMGH);
                    const int xc = min(max(xi, 0), IMGW - 1);
                    const int yc = min(max(yi, 0), IMGH - 1);
                    const float cw = wx * wy * (valid ? 1.f : 0.f);
                    const float v = img[((size_t)(yc * IMGW + xc)) * HD + lane];
                    s = fmaf(cw, v, s);
                }
            }
            acc = fmaf(wts[p], s, acc);
        }
    }

    const size_t oi = ((size_t)b * NQ + q) * C + h * HD + lane;
    out[oi] = query[oi] + 0.5f * acc;   // mean over the 2 bev queues
}

// ---------------------------------------------------------------------------
// Host-side launcher
// ---------------------------------------------------------------------------
extern "C" void kernel_launch(void* const* d_in, const int* in_sizes, int n_in,
                              void* d_out, int out_size, void* d_ws, size_t ws_size,
                              hipStream_t stream) {
    const float* query     = (const float*)d_in[0];
    const float* query_pos = (const float*)d_in[1];
    const float* refpts    = (const float*)d_in[2];
    // d_in[3] = spatial_shapes (int32, compile-time 128x128 here)
    const float* Wv        = (const float*)d_in[4];
    const float* bv        = (const float*)d_in[5];
    const float* Wo        = (const float*)d_in[6];
    const float* bo        = (const float*)d_in[7];
    const float* Wa        = (const float*)d_in[8];
    const float* ba        = (const float*)d_in[9];
    float* out             = (float*)d_out;

    // Workspace: value images then raw off/aw.
    float* vimg  = (float*)d_ws;                                  // 32 MiB
    float* offaw = vimg + (size_t)BS * NH * NQ * HD;              // 24 MiB

    gemm_value_kernel<<<dim3(NQ / 32, 2, BS), 256, 0, stream>>>(
        query, Wv, bv, vimg);
    gemm_offaw_kernel<<<dim3(NQ / 32, BS), 384, 0, stream>>>(
        query, query_pos, Wo, bo, Wa, ba, offaw);
    sample_kernel<<<(BS * NQ * NH) / 8, 256, 0, stream>>>(
        query, refpts, vimg, offaw, out);
}